// DecodePredictions_23459111371191
// MI455X (gfx1250) — compile-verified
//
#include <hip/hip_runtime.h>
#include <hip/hip_bf16.h>
#include <stdint.h>

// ---------------- problem constants (from reference) ----------------
#define B_BATCH   8
#define A_TOTAL   49104          // anchors over FPN levels 3..7 at 512x512
#define C_CLS     80
#define K_TOP     100            // MAX_PER_CLASS
#define MAX_DET   100
#define CONF_T    0.05f
#define IOU_T     0.5f

#define CAND      1024           // per-class candidate buffer (pow2, bitonic)
#define FIN_N     (C_CLS * K_TOP)   // 8000
#define FIN_NS    8192              // pow2 pad for final bitonic

// dynamic LDS sizes
#define COL_BYTES ((size_t)A_TOTAL * 4)                       // 196416
#define FIN_BYTES ((size_t)FIN_NS * 8 + (size_t)FIN_N * 4)    // 97536

// ---------------- helpers ----------------
__device__ __forceinline__ unsigned fkey(float f) {
  unsigned u = __float_as_uint(f);
  return (u & 0x80000000u) ? ~u : (u | 0x80000000u);   // monotonic: bigger float -> bigger key
}
__device__ __forceinline__ float fkey_inv(unsigned u) {
  unsigned v = (u & 0x80000000u) ? (u & 0x7FFFFFFFu) : ~u;
  return __uint_as_float(v);
}

// CDNA5 async global->LDS (ASYNCcnt path).  lds_off = byte offset of LDS dest.
__device__ __forceinline__ void async_g2l_b32(unsigned lds_off, unsigned long long gaddr) {
  asm volatile("global_load_async_to_lds_b32 %0, %1, off" :: "v"(lds_off), "v"(gaddr) : "memory");
}
__device__ __forceinline__ void async_g2l_b128(unsigned lds_off, unsigned long long gaddr) {
  asm volatile("global_load_async_to_lds_b128 %0, %1, off" :: "v"(lds_off), "v"(gaddr) : "memory");
}
__device__ __forceinline__ void wait_async0() {
  asm volatile("s_wait_asynccnt 0" ::: "memory");
}
__device__ __forceinline__ unsigned lds_off_of(const void* p) {
  return (unsigned)(uintptr_t)p;   // low 32 bits of a generic LDS pointer = LDS byte offset
}

// ---------------- kernel 1: box decode (streaming, anchors analytic) ----------------
__global__ __launch_bounds__(256) void decode_boxes_kernel(
    const float* __restrict__ box_pred, float* __restrict__ out_boxes) {
  int i = blockIdx.x * blockDim.x + threadIdx.x;
  if (i >= B_BATCH * A_TOTAL) return;
  int a = i % A_TOTAL;

  // level boundaries: 36864, 9216, 2304, 576, 144 (9 anchors/cell)
  int lvl = 0, start = 0;
  if      (a >= 48960) { lvl = 4; start = 48960; }
  else if (a >= 48384) { lvl = 3; start = 48384; }
  else if (a >= 46080) { lvl = 2; start = 46080; }
  else if (a >= 36864) { lvl = 1; start = 36864; }
  int idx  = a - start;
  int d    = idx % 9;
  int cell = idx / 9;
  int fw   = 64 >> lvl;
  int x    = cell % fw;
  int y    = cell / fw;
  float stride = (float)(8 << lvl);
  float acx = ((float)x + 0.5f) * stride;
  float acy = ((float)y + 0.5f) * stride;
  float sa  = (float)(32 << lvl);              // sqrt(area)
  const float sqr[3] = {0.70710678118654752f, 1.0f, 1.41421356237309505f}; // sqrt(ratio)
  const float scl[3] = {1.0f, 1.2599210498948732f, 1.5874010519681994f};   // 2^(k/3)
  int ri = d / 3, si = d % 3;
  float aw = sa * sqr[ri] * scl[si];           // aw = sqrt(area*r)*s
  float ah = sa / sqr[ri] * scl[si];           // ah = sqrt(area/r)*s

  float4 p = ((const float4*)box_pred)[i];
  float bx = p.x * 0.1f, by = p.y * 0.1f, bw = p.z * 0.2f, bh = p.w * 0.2f;
  float ncx = bx * aw + acx;
  float ncy = by * ah + acy;
  float nw  = expf(bw) * aw;
  float nh  = expf(bh) * ah;
  float4 o;
  o.x = ncx - nw * 0.5f; o.y = ncy - nh * 0.5f;
  o.z = ncx + nw * 0.5f; o.w = ncy + nh * 0.5f;
  ((float4*)out_boxes)[i] = o;
}

// ---------------- kernel 2: per-(batch,class) exact top-100 + NMS ----------------
// One workgroup per (b,c).  Whole class column staged into LDS via async DMA,
// then 3-round radix select + bitonic sort + greedy NMS, all LDS-resident.
__global__ __launch_bounds__(256) void select_nms_kernel(
    const float* __restrict__ cls_pred,   // [B, A, C]
    const float* __restrict__ boxes,      // [B, A, 4] xyxy
    float* __restrict__ cls_boxes,        // [B*C, K, 4]
    float* __restrict__ cls_scores) {     // [B*C, K]
  extern __shared__ unsigned char dynsmem[];
  float* s_col = (float*)dynsmem;                       // A_TOTAL floats (196416 B)

  __shared__ unsigned s_hist[256];
  __shared__ unsigned long long s_cand[CAND];
  __shared__ unsigned s_cnt, s_sel, s_above;
  __shared__ __align__(16) float s_nbox[K_TOP * 4];
  __shared__ float s_nsc[K_TOP];
  __shared__ int   s_keep[K_TOP];

  const int tid = threadIdx.x;
  const int bc  = blockIdx.x;
  const int b   = bc / C_CLS;
  const int c   = bc % C_CLS;

  // ---- stage class column (stride C floats) into LDS with async DMA ----
  const float* src = cls_pred + (size_t)b * A_TOTAL * C_CLS + c;
  for (int a = tid; a < A_TOTAL; a += 256) {
    async_g2l_b32(lds_off_of(&s_col[a]),
                  (unsigned long long)(uintptr_t)(src + (size_t)a * C_CLS));
  }
  wait_async0();
  __syncthreads();

  // ---- 3-round 8-bit radix select for the 100th largest key ----
  unsigned prefix = 0, remaining = K_TOP;
  for (int shift = 24; shift >= 8; shift -= 8) {
    s_hist[tid] = 0;
    __syncthreads();
    for (int a = tid; a < A_TOTAL; a += 256) {
      unsigned u = fkey(s_col[a]);
      unsigned hi = (u >> shift) >> 8;          // bits above current digit
      if (hi == prefix) atomicAdd(&s_hist[(u >> shift) & 0xFF], 1u);
    }
    __syncthreads();
    if (tid == 0) {
      unsigned cum = 0, sel = 0, above = 0;
      for (int dd = 255; dd >= 0; --dd) {
        unsigned h = s_hist[dd];
        if (cum + h >= remaining) { sel = (unsigned)dd; above = cum; break; }
        cum += h;
      }
      s_sel = sel; s_above = above;
    }
    __syncthreads();
    remaining -= s_above;
    prefix = (prefix << 8) | s_sel;
    __syncthreads();
  }

  // ---- compact candidates: key top-24 bits >= prefix ----
  if (tid == 0) s_cnt = 0;
  __syncthreads();
  for (int a = tid; a < A_TOTAL; a += 256) {
    unsigned u = fkey(s_col[a]);
    if ((u >> 8) >= prefix) {
      unsigned p = atomicAdd(&s_cnt, 1u);
      if (p < CAND)
        s_cand[p] = ((unsigned long long)u << 32) | (unsigned)(0xFFFFFFFFu - (unsigned)a);
    }
  }
  __syncthreads();
  unsigned stored = s_cnt < CAND ? s_cnt : CAND;
  for (int i = tid; i < CAND; i += 256)
    if ((unsigned)i >= stored) s_cand[i] = 0ull;       // sentinel: smallest
  __syncthreads();

  // ---- bitonic sort ascending; top-100 at the tail ----
  for (unsigned k = 2; k <= CAND; k <<= 1) {
    for (unsigned j = k >> 1; j > 0; j >>= 1) {
      for (unsigned i = tid; i < CAND; i += 256) {
        unsigned ixj = i ^ j;
        if (ixj > i) {
          bool asc = ((i & k) == 0);
          unsigned long long xv = s_cand[i], yv = s_cand[ixj];
          if ((xv > yv) == asc) { s_cand[i] = yv; s_cand[ixj] = xv; }
        }
      }
      __syncthreads();
    }
  }

  // ---- extract top-100, async-DMA their boxes into LDS ----
  if (tid < K_TOP) {
    unsigned long long e = s_cand[CAND - 1 - tid];
    unsigned u = (unsigned)(e >> 32);
    unsigned a = 0xFFFFFFFFu - (unsigned)e;
    if (a >= A_TOTAL) a = 0;                           // padding safety
    float sc = fkey_inv(u);
    s_nsc[tid]  = sc;
    s_keep[tid] = (sc > CONF_T) ? 1 : 0;
    async_g2l_b128(lds_off_of(&s_nbox[tid * 4]),
                   (unsigned long long)(uintptr_t)(boxes + ((size_t)b * A_TOTAL + a) * 4));
  }
  wait_async0();
  __syncthreads();

  // ---- greedy NMS over 100 boxes (sequential suppressor, parallel victims) ----
  for (int i = 0; i < K_TOP - 1; ++i) {
    __syncthreads();
    if (tid > i && tid < K_TOP && s_keep[i]) {
      const float* A4 = &s_nbox[i * 4];
      const float* B4 = &s_nbox[tid * 4];
      float areaA = (A4[2] - A4[0]) * (A4[3] - A4[1]);
      float areaB = (B4[2] - B4[0]) * (B4[3] - B4[1]);
      float lx = fmaxf(A4[0], B4[0]), ly = fmaxf(A4[1], B4[1]);
      float rx = fminf(A4[2], B4[2]), ry = fminf(A4[3], B4[3]);
      float w = fmaxf(rx - lx, 0.0f), h = fmaxf(ry - ly, 0.0f);
      float inter = w * h;
      float iou = inter / (areaA + areaB - inter + 1e-8f);
      if (iou > IOU_T) s_keep[tid] = 0;
    }
  }
  __syncthreads();

  // ---- emit per-class results ----
  if (tid < K_TOP) {
    cls_scores[(size_t)bc * K_TOP + tid] = s_keep[tid] ? s_nsc[tid] : -1.0f;
    float4 o;
    o.x = s_nbox[tid * 4 + 0]; o.y = s_nbox[tid * 4 + 1];
    o.z = s_nbox[tid * 4 + 2]; o.w = s_nbox[tid * 4 + 3];
    ((float4*)cls_boxes)[(size_t)bc * K_TOP + tid] = o;
  }
}

// ---------------- kernel 3: per-batch merge (top-100 of 8000) ----------------
__global__ __launch_bounds__(256) void final_topk_kernel(
    const float* __restrict__ cls_scores,   // [B, C*K]
    const float* __restrict__ cls_boxes,    // [B, C*K, 4]
    float* __restrict__ out) {              // [B,100,4] ++ [B,100] ++ [B,100]
  extern __shared__ unsigned char dynsmem[];
  unsigned long long* cand = (unsigned long long*)dynsmem;          // FIN_NS * 8
  float* s_sc = (float*)(dynsmem + (size_t)FIN_NS * 8);             // FIN_N floats

  const int tid = threadIdx.x;
  const int b   = blockIdx.x;

  // async-DMA the 8000 masked scores into LDS (b128, fully coalesced)
  const float* src = cls_scores + (size_t)b * FIN_N;
  for (int t = tid; t < FIN_N / 4; t += 256) {
    async_g2l_b128(lds_off_of(&s_sc[t * 4]),
                   (unsigned long long)(uintptr_t)(src + (size_t)t * 4));
  }
  wait_async0();
  __syncthreads();

  for (int i = tid; i < FIN_NS; i += 256) {
    if (i < FIN_N) {
      unsigned u = fkey(s_sc[i]);
      cand[i] = ((unsigned long long)u << 32) | (unsigned)(0xFFFFFFFFu - (unsigned)i);
    } else {
      cand[i] = 0ull;
    }
  }
  __syncthreads();

  for (unsigned k = 2; k <= FIN_NS; k <<= 1) {
    for (unsigned j = k >> 1; j > 0; j >>= 1) {
      for (unsigned i = tid; i < FIN_NS; i += 256) {
        unsigned ixj = i ^ j;
        if (ixj > i) {
          bool asc = ((i & k) == 0);
          unsigned long long xv = cand[i], yv = cand[ixj];
          if ((xv > yv) == asc) { cand[i] = yv; cand[ixj] = xv; }
        }
      }
      __syncthreads();
    }
  }

  if (tid < MAX_DET) {
    unsigned long long e = cand[FIN_NS - 1 - tid];
    unsigned u = (unsigned)(e >> 32);
    unsigned i = 0xFFFFFFFFu - (unsigned)e;
    float fs = fkey_inv(u);
    bool valid = (i < (unsigned)FIN_N) && (fs > CONF_T);
    float4 bx = make_float4(0.f, 0.f, 0.f, 0.f);
    float cls = -1.0f, sc = 0.0f;
    if (valid) {
      bx  = ((const float4*)cls_boxes)[(size_t)b * FIN_N + i];
      cls = (float)(i / K_TOP);
      sc  = fs;
    }
    float* out_b = out;                                   // [B,100,4]
    float* out_s = out + (size_t)B_BATCH * MAX_DET * 4;   // [B,100]
    float* out_c = out_s + (size_t)B_BATCH * MAX_DET;     // [B,100]
    ((float4*)out_b)[(size_t)b * MAX_DET + tid] = bx;
    out_s[(size_t)b * MAX_DET + tid] = sc;
    out_c[(size_t)b * MAX_DET + tid] = cls;
  }
}

// ---------------- host entry ----------------
extern "C" void kernel_launch(void* const* d_in, const int* in_sizes, int n_in,
                              void* d_out, int out_size, void* d_ws, size_t ws_size,
                              hipStream_t stream) {
  (void)in_sizes; (void)n_in; (void)out_size; (void)ws_size;
  const float* box_pred = (const float*)d_in[1];   // [8,49104,4]
  const float* cls_pred = (const float*)d_in[2];   // [8,49104,80]
  float* ws = (float*)d_ws;

  // scratch layout (floats): boxes | per-class boxes | per-class scores
  float* ws_boxes      = ws;                                           // B*A*4 = 1,571,328
  float* ws_cls_boxes  = ws + (size_t)B_BATCH * A_TOTAL * 4;           // B*C*K*4 = 256,000
  float* ws_cls_scores = ws_cls_boxes + (size_t)B_BATCH * C_CLS * K_TOP * 4; // B*C*K = 64,000

  // allow >default dynamic LDS (CDNA5 has 320 KB / WGP)
  (void)hipFuncSetAttribute((const void*)select_nms_kernel,
                            hipFuncAttributeMaxDynamicSharedMemorySize, (int)COL_BYTES);
  (void)hipFuncSetAttribute((const void*)final_topk_kernel,
                            hipFuncAttributeMaxDynamicSharedMemorySize, (int)FIN_BYTES);

  int n1 = B_BATCH * A_TOTAL;
  decode_boxes_kernel<<<(n1 + 255) / 256, 256, 0, stream>>>(box_pred, ws_boxes);

  select_nms_kernel<<<B_BATCH * C_CLS, 256, (size_t)COL_BYTES, stream>>>(
      cls_pred, ws_boxes, ws_cls_boxes, ws_cls_scores);

  final_topk_kernel<<<B_BATCH, 256, (size_t)FIN_BYTES, stream>>>(
      ws_cls_scores, ws_cls_boxes, (float*)d_out);
}